// GraphNet_12807592477363
// MI455X (gfx1250) — compile-verified
//
#include <hip/hip_runtime.h>
#include <hip/hip_bf16.h>
#include <math.h>

// Problem constants (match reference)
#define G_    3
#define N_    200000
#define E_    3200000
#define D_    128
#define H_    16
#define BN_   2048
#define NCLS_ 2
#define NHID_ 8

typedef __attribute__((ext_vector_type(2))) float v2f;
typedef __attribute__((ext_vector_type(8))) float v8f;

// ---------------------------------------------------------------------------
// Workspace layout (floats)
// ---------------------------------------------------------------------------
#define OFF_H      ((size_t)0)
#define OFF_ASRC   (OFF_H    + (size_t)G_*N_*H_)
#define OFF_ADST   (OFF_ASRC + (size_t)G_*N_)
#define OFF_M      (OFF_ADST + (size_t)G_*N_)
#define OFF_DEN    (OFF_M    + (size_t)G_*N_)
#define OFF_AGG    (OFF_DEN  + (size_t)G_*N_)
#define OFF_PSUM   (OFF_AGG  + (size_t)G_*N_*H_)
#define OFF_PCNT   (OFF_PSUM + (size_t)G_*BN_)

__device__ __forceinline__ float leaky2(float v) { return v > 0.f ? v : 0.2f * v; }

__device__ __forceinline__ float atomicMaxFloat(float* addr, float val) {
    if (val >= 0.f)
        return __int_as_float(atomicMax((int*)addr, __float_as_int(val)));
    else
        return __uint_as_float(atomicMin((unsigned int*)addr, __float_as_uint(val)));
}

// ---------------------------------------------------------------------------
// Kernel 0: initialize accumulators (runs every launch; graph-replayed)
// ---------------------------------------------------------------------------
__global__ void __launch_bounds__(256) k_init(float* m, float* denom,
                                              float* agg, float* pooled /* psum|pcnt */) {
    size_t idx = (size_t)blockIdx.x * blockDim.x + threadIdx.x;
    if (idx < (size_t)G_ * N_) {
        m[idx]     = -__builtin_inff();
        denom[idx] = 0.f;
        float* o = agg + idx * H_;
        #pragma unroll
        for (int j = 0; j < H_; ++j) o[j] = 0.f;
    }
    if (idx < (size_t)G_ * BN_ * 2) pooled[idx] = 0.f;
}

// ---------------------------------------------------------------------------
// Kernel 1: h = x @ W_gat (f32 WMMA), fused asrc/adst = h . att_{src,dst}
//   block = 256 threads = 8 waves; wave w owns a 16-row tile; 128 rows/block
//   B fragments hoisted to registers: steady-state loop = NT-load(A) + wmma.
// ---------------------------------------------------------------------------
__global__ void __launch_bounds__(256) k_gat_linear(
    const float* __restrict__ x, const float* __restrict__ Wgat,
    const float* __restrict__ att_src, const float* __restrict__ att_dst,
    float* __restrict__ h, float* __restrict__ asrc, float* __restrict__ adst) {

    __shared__ float sW[D_ * H_];        // 128x16 weights (8 KB)
    __shared__ float sT[8 * 16 * H_];    // 8 waves x 16x16 output tiles (8 KB)
    __shared__ float sAs[H_], sAd[H_];

    const int g = blockIdx.y;
    const float* xg  = x    + (size_t)g * N_ * D_;
    const float* Wg  = Wgat + (size_t)g * D_ * H_;
    float*       hg  = h    + (size_t)g * N_ * H_;

    const int t = threadIdx.x;
    for (int i = t; i < D_ * H_; i += 256) sW[i] = Wg[i];
    if (t < H_) { sAs[t] = att_src[g * H_ + t]; sAd[t] = att_dst[g * H_ + t]; }
    __syncthreads();

    const int wave = t >> 5;
    const int lane = t & 31;
    const int col  = lane & 15;     // N index (B/C/D) and M index (A)
    const int hi   = lane >> 4;     // lane half
    const int kb   = hi * 2;        // K offset within a 4-wide step

    // Preload ALL B fragments for K=0..127 into registers (64 VGPRs).
    // B: 4x16 f32 tile per step; lane holds N=col, {K=4s+kb, K=4s+kb+1}.
    v2f breg[D_ / 4];
    #pragma unroll
    for (int s = 0; s < D_ / 4; ++s) {
        const int k = 4 * s + kb;
        breg[s].x = sW[k * H_ + col];
        breg[s].y = sW[(k + 1) * H_ + col];
    }

    const int rowBase = blockIdx.x * 128 + wave * 16;
    const int row     = rowBase + col;
    const int rowc    = row < N_ ? row : N_ - 1;      // clamp (branchless, EXEC stays full)
    const float* xr   = xg + (size_t)rowc * D_;

    v8f acc = {};
    #pragma unroll
    for (int s = 0; s < D_ / 4; ++s) {
        const int k = 4 * s + kb;
        // A: 16x4 f32 tile; lane holds M=col, {K=k, K=k+1}. Streamed once -> NT.
        v2f a = __builtin_nontemporal_load(reinterpret_cast<const v2f*>(xr + k));
        acc = __builtin_amdgcn_wmma_f32_16x16x4_f32(
            /*neg_a=*/false, a, /*neg_b=*/false, breg[s],
            /*c_mod=*/(short)0, acc, /*reuse_a=*/false, /*reuse_b=*/false);
    }

    // D layout: VGPR r -> (lanes 0-15: M=r, N=lane) (lanes 16-31: M=r+8, N=lane-16)
    const int mbase = hi * 8;
    float* sTile = sT + wave * 256;
    #pragma unroll
    for (int r = 0; r < 8; ++r) {
        const int mrow = rowBase + mbase + r;
        sTile[(mbase + r) * H_ + col] = acc[r];
        if (mrow < N_) hg[(size_t)mrow * H_ + col] = acc[r];
    }
    __syncthreads();

    // Per-node attention dot products from the LDS tiles (threads 0..127)
    if (t < 128) {
        const int n = blockIdx.x * 128 + t;
        if (n < N_) {
            const float* hv = sT + t * H_;
            float s1 = 0.f, s2 = 0.f;
            #pragma unroll
            for (int j = 0; j < H_; ++j) { float v = hv[j]; s1 += v * sAs[j]; s2 += v * sAd[j]; }
            asrc[(size_t)g * N_ + n] = s1;
            adst[(size_t)g * N_ + n] = s2;
        }
    }
}

// ---------------------------------------------------------------------------
// Edge passes (edges [0,E) plus self-loops [E, E+N))
// ---------------------------------------------------------------------------
__global__ void __launch_bounds__(256) k_edge_max(
    const int* __restrict__ ei, const float* __restrict__ asrc,
    const float* __restrict__ adst, float* __restrict__ m) {
    const int g = blockIdx.y;
    const size_t idx = (size_t)blockIdx.x * blockDim.x + threadIdx.x;
    int s, d;
    if (idx < (size_t)E_) {
        const int* eg = ei + (size_t)g * 2 * E_;
        s = eg[idx]; d = eg[E_ + idx];
    } else if (idx < (size_t)(E_ + N_)) {
        s = d = (int)(idx - E_);
    } else return;
    const float v = leaky2(asrc[(size_t)g * N_ + s] + adst[(size_t)g * N_ + d]);
    atomicMaxFloat(&m[(size_t)g * N_ + d], v);
}

__global__ void __launch_bounds__(256) k_edge_sum(
    const int* __restrict__ ei, const float* __restrict__ asrc,
    const float* __restrict__ adst, const float* __restrict__ m,
    float* __restrict__ denom) {
    const int g = blockIdx.y;
    const size_t idx = (size_t)blockIdx.x * blockDim.x + threadIdx.x;
    int s, d;
    if (idx < (size_t)E_) {
        const int* eg = ei + (size_t)g * 2 * E_;
        s = eg[idx]; d = eg[E_ + idx];
    } else if (idx < (size_t)(E_ + N_)) {
        s = d = (int)(idx - E_);
    } else return;
    const float v = leaky2(asrc[(size_t)g * N_ + s] + adst[(size_t)g * N_ + d]);
    const float w = __expf(v - m[(size_t)g * N_ + d]);
    atomicAdd(&denom[(size_t)g * N_ + d], w);
}

__global__ void __launch_bounds__(256) k_edge_agg(
    const int* __restrict__ ei, const float* __restrict__ asrc,
    const float* __restrict__ adst, const float* __restrict__ m,
    const float* __restrict__ denom, const float* __restrict__ h,
    float* __restrict__ agg) {
    const int g = blockIdx.y;
    const size_t idx = (size_t)blockIdx.x * blockDim.x + threadIdx.x;
    int s, d;
    if (idx < (size_t)E_) {
        const int* eg = ei + (size_t)g * 2 * E_;
        s = eg[idx]; d = eg[E_ + idx];
    } else if (idx < (size_t)(E_ + N_)) {
        s = d = (int)(idx - E_);
    } else return;
    const size_t gn = (size_t)g * N_;
    const float v = leaky2(asrc[gn + s] + adst[gn + d]);
    const float alpha = __expf(v - m[gn + d]) / denom[gn + d];
    const float* hs = h + (gn + s) * H_;
    float* od = agg + (gn + d) * H_;
    #pragma unroll
    for (int j = 0; j < H_; ++j) atomicAdd(&od[j], alpha * hs[j]);
}

// ---------------------------------------------------------------------------
// Kernel 5: post-linear (H->1) + global mean pool accumulation
// ---------------------------------------------------------------------------
__global__ void __launch_bounds__(256) k_post_pool(
    const float* __restrict__ agg, const float* __restrict__ b_gat,
    const float* __restrict__ W_post, const float* __restrict__ b_post,
    const int* __restrict__ batch, float* __restrict__ psum, float* __restrict__ pcnt) {
    const int g = blockIdx.y;
    const int n = blockIdx.x * blockDim.x + threadIdx.x;
    if (n >= N_) return;
    const float* o = agg + ((size_t)g * N_ + n) * H_;
    float v = b_post[g];
    #pragma unroll
    for (int j = 0; j < H_; ++j) v += (o[j] + b_gat[g * H_ + j]) * W_post[g * H_ + j];
    const int b = batch[(size_t)g * N_ + n];
    atomicAdd(&psum[g * BN_ + b], v);
    atomicAdd(&pcnt[g * BN_ + b], 1.0f);
}

// ---------------------------------------------------------------------------
// Kernel 6: MLP head on pooled [BN, G] -> [BN, 2]
// ---------------------------------------------------------------------------
__global__ void __launch_bounds__(256) k_mlp(
    const float* __restrict__ psum, const float* __restrict__ pcnt,
    const float* __restrict__ W0, const float* __restrict__ b0,
    const float* __restrict__ Wh, const float* __restrict__ bh,
    const float* __restrict__ Wf, const float* __restrict__ bf,
    float* __restrict__ outp) {
    const int r = blockIdx.x * blockDim.x + threadIdx.x;
    if (r >= BN_) return;

    float pg[G_];
    #pragma unroll
    for (int g = 0; g < G_; ++g)
        pg[g] = psum[g * BN_ + r] / fmaxf(pcnt[g * BN_ + r], 1.0f);

    float p[H_];
    #pragma unroll
    for (int j = 0; j < H_; ++j) {
        float v = b0[j];
        #pragma unroll
        for (int g = 0; g < G_; ++g) v += pg[g] * W0[g * H_ + j];
        p[j] = v > 0.f ? v : 0.01f * v;
    }
    for (int l = 0; l < NHID_; ++l) {
        float q[H_];
        #pragma unroll
        for (int j = 0; j < H_; ++j) {
            float v = bh[l * H_ + j];
            #pragma unroll
            for (int k = 0; k < H_; ++k) v += p[k] * Wh[((size_t)l * H_ + k) * H_ + j];
            q[j] = v > 0.f ? v : 0.01f * v;
        }
        #pragma unroll
        for (int j = 0; j < H_; ++j) p[j] = q[j];
    }
    #pragma unroll
    for (int c = 0; c < NCLS_; ++c) {
        float v = bf[c];
        #pragma unroll
        for (int k = 0; k < H_; ++k) v += p[k] * Wf[k * NCLS_ + c];
        outp[r * NCLS_ + c] = v;
    }
}

// ---------------------------------------------------------------------------
extern "C" void kernel_launch(void* const* d_in, const int* in_sizes, int n_in,
                              void* d_out, int out_size, void* d_ws, size_t ws_size,
                              hipStream_t stream) {
    const float* x        = (const float*)d_in[0];
    const int*   ei       = (const int*)  d_in[1];
    const int*   batch    = (const int*)  d_in[2];
    // d_in[3] = B (scalar 2048, hardcoded as BN_)
    const float* W_gat    = (const float*)d_in[4];
    const float* att_src  = (const float*)d_in[5];
    const float* att_dst  = (const float*)d_in[6];
    const float* b_gat    = (const float*)d_in[7];
    const float* W_post   = (const float*)d_in[8];
    const float* b_post   = (const float*)d_in[9];
    const float* W0       = (const float*)d_in[10];
    const float* b0       = (const float*)d_in[11];
    const float* Wh       = (const float*)d_in[12];
    const float* bh       = (const float*)d_in[13];
    const float* Wf       = (const float*)d_in[14];
    const float* bf       = (const float*)d_in[15];
    float* out = (float*)d_out;

    float* ws    = (float*)d_ws;
    float* h     = ws + OFF_H;
    float* asrc  = ws + OFF_ASRC;
    float* adst  = ws + OFF_ADST;
    float* m     = ws + OFF_M;
    float* denom = ws + OFF_DEN;
    float* agg   = ws + OFF_AGG;
    float* psum  = ws + OFF_PSUM;
    float* pcnt  = ws + OFF_PCNT;

    // 0) init accumulators
    {
        int total = G_ * N_;
        k_init<<<(total + 255) / 256, 256, 0, stream>>>(m, denom, agg, psum);
    }
    // 1) WMMA GEMM + attention dots
    {
        dim3 grid((N_ + 127) / 128, G_);
        k_gat_linear<<<grid, 256, 0, stream>>>(x, W_gat, att_src, att_dst, h, asrc, adst);
    }
    // 2-4) edge softmax passes
    {
        dim3 grid((E_ + N_ + 255) / 256, G_);
        k_edge_max<<<grid, 256, 0, stream>>>(ei, asrc, adst, m);
        k_edge_sum<<<grid, 256, 0, stream>>>(ei, asrc, adst, m, denom);
        k_edge_agg<<<grid, 256, 0, stream>>>(ei, asrc, adst, m, denom, h, agg);
    }
    // 5) post-linear + pooling
    {
        dim3 grid((N_ + 255) / 256, G_);
        k_post_pool<<<grid, 256, 0, stream>>>(agg, b_gat, W_post, b_post, batch, psum, pcnt);
    }
    // 6) MLP head
    k_mlp<<<(BN_ + 255) / 256, 256, 0, stream>>>(psum, pcnt, W0, b0, Wh, bh, Wf, bf, out);
}